// ReservoirSampler_10711648436601
// MI455X (gfx1250) — compile-verified
//
#include <hip/hip_runtime.h>
#include <stdint.h>

// Reservoir "parallel add": (1) replicate JAX threefry randint to find, per
// slot, the LAST sample that targeted it; (2) bandwidth-bound row gather
// (~128 MiB total traffic -> ~6 us at 23.3 TB/s) done via the CDNA5 async
// global<->LDS DMA path (ASYNCcnt) instead of VGPR round-trips.

#define D_FEAT 1024
#define ROW_THREADS 256   // 256 lanes * 16B = 4096B = one row per block

// ---------------------------------------------------------------------------
// Threefry2x32 with key = (0, 42)  (jax.random.key(42) -> [seed>>32, seed&~0])
// ---------------------------------------------------------------------------
__device__ __forceinline__ void threefry2x32_k42(uint32_t x0, uint32_t x1,
                                                 uint32_t& o0, uint32_t& o1) {
  const uint32_t ks0 = 0u;
  const uint32_t ks1 = 42u;
  const uint32_t ks2 = ks0 ^ ks1 ^ 0x1BD11BDAu;  // = 42 ^ 0x1BD11BDA
  x0 += ks0;
  x1 += ks1;
#define TF_ROUND(r)                                   \
  do {                                                \
    x0 += x1;                                         \
    x1 = (x1 << (r)) | (x1 >> (32 - (r)));            \
    x1 ^= x0;                                         \
  } while (0)
  // group 0 (rot 13,15,26,6), inject ks1/ks2+1
  TF_ROUND(13); TF_ROUND(15); TF_ROUND(26); TF_ROUND(6);
  x0 += ks1; x1 += ks2 + 1u;
  // group 1 (rot 17,29,16,24), inject ks2/ks0+2
  TF_ROUND(17); TF_ROUND(29); TF_ROUND(16); TF_ROUND(24);
  x0 += ks2; x1 += ks0 + 2u;
  // group 2
  TF_ROUND(13); TF_ROUND(15); TF_ROUND(26); TF_ROUND(6);
  x0 += ks0; x1 += ks1 + 3u;
  // group 3
  TF_ROUND(17); TF_ROUND(29); TF_ROUND(16); TF_ROUND(24);
  x0 += ks1; x1 += ks2 + 4u;
  // group 4
  TF_ROUND(13); TF_ROUND(15); TF_ROUND(26); TF_ROUND(6);
  x0 += ks2; x1 += ks0 + 5u;
#undef TF_ROUND
  o0 = x0;
  o1 = x1;
}

// ---------------------------------------------------------------------------
// Pass 0: winner[s] = -1
// ---------------------------------------------------------------------------
__global__ void rs_winner_init(int* __restrict__ winner, int n) {
  int i = blockIdx.x * blockDim.x + threadIdx.x;
  if (i < n) winner[i] = -1;
}

// ---------------------------------------------------------------------------
// Pass 1: per-sample target slot + atomicMax(order).
// JAX randint(key, (N,), 0, idx+1):
//   bits = threefry_2x32(key, iota(2N)) -> block k uses counters (k, k+N),
//   hi = x0-lane, lo = x1-lane;
//   span = idx+1; mult = ((2^16 % span)^2) % span  (== 2^32 % span);
//   j = (hi*mult + lo)  [wrapping u32]  % span.
// ---------------------------------------------------------------------------
__global__ void rs_winner_compute(const int* __restrict__ i0p,
                                  int* __restrict__ winner, int N, int n) {
  int k = blockIdx.x * blockDim.x + threadIdx.x;
  if (k >= N) return;
  const int i0 = i0p[0];
  const int idx = i0 + k;
  int target;
  if (idx < n) {
    target = idx;  // fill phase: deterministic slot
  } else {
    const uint32_t span = (uint32_t)idx + 1u;
    uint32_t hi, lo;
    threefry2x32_k42((uint32_t)k, (uint32_t)(k + N), hi, lo);
    uint32_t mult = 65536u % span;        // 2^16 % span
    mult = (mult * mult) % span;          // 2^32 % span (fits: 65535^2 < 2^32)
    const uint32_t j = (hi * mult + lo) % span;  // wrapping u32, like lax
    if (j >= (uint32_t)n) return;         // discarded sample
    target = (int)j;
  }
  atomicMax(&winner[target], k);          // last (max order) writer wins
}

// ---------------------------------------------------------------------------
// Pass 2: row gather via CDNA5 async DMA.
//   global_load_async_to_lds_b128  : HBM row -> LDS tile   (ASYNCcnt)
//   s_wait_asynccnt 0
//   global_store_async_from_lds_b128: LDS tile -> HBM out  (ASYNCcnt)
// LDS byte offset = low 32 bits of the flat pointer (LDS aperture occupies
// addr[63:32] only, ISA 10.2).
// ---------------------------------------------------------------------------
__global__ void __launch_bounds__(ROW_THREADS)
rs_gather_async(const float* __restrict__ buffer,
                const float* __restrict__ samples,
                const int* __restrict__ winner,
                float* __restrict__ out) {
  __shared__ __align__(16) float tile[D_FEAT];

  const int row = blockIdx.x;
  const int w = winner[row];  // uniform per block
  const float* src = (w >= 0) ? (samples + (size_t)w * D_FEAT)
                              : (buffer + (size_t)row * D_FEAT);

  const int t = threadIdx.x;                 // 0..255, 16B per lane
  const uint64_t gsrc =
      (uint64_t)(uintptr_t)(src + ((size_t)t << 2));
  const uint64_t gdst =
      (uint64_t)(uintptr_t)(out + (size_t)row * D_FEAT + ((size_t)t << 2));
  const uint32_t lds_off =
      (uint32_t)(uintptr_t)(void*)&tile[t << 2];  // flat -> LDS byte offset

  // HBM -> LDS (async unit, 16B/lane)
  asm volatile("global_load_async_to_lds_b128 %0, %1, off"
               :: "v"(lds_off), "v"(gsrc)
               : "memory");
  // our own wave's async load must land before we stream it back out
  asm volatile("s_wait_asynccnt 0" ::: "memory");
  // LDS -> HBM (async unit)
  asm volatile("global_store_async_from_lds_b128 %0, %1, off"
               :: "v"(gdst), "v"(lds_off)
               : "memory");
  // drain before LDS dealloc at wave end (S_ENDPGM also wait-idles)
  asm volatile("s_wait_asynccnt 0" ::: "memory");
}

// ---------------------------------------------------------------------------
extern "C" void kernel_launch(void* const* d_in, const int* in_sizes, int n_in,
                              void* d_out, int out_size, void* d_ws,
                              size_t ws_size, hipStream_t stream) {
  const float* buffer  = (const float*)d_in[0];
  const float* samples = (const float*)d_in[1];
  const int*   i0p     = (const int*)d_in[2];
  float* out = (float*)d_out;

  const int n = in_sizes[0] / D_FEAT;  // 16384
  const int N = in_sizes[1] / D_FEAT;  // 65536

  int* winner = (int*)d_ws;  // n ints of scratch (64 KiB)

  rs_winner_init<<<(n + 255) / 256, 256, 0, stream>>>(winner, n);
  rs_winner_compute<<<(N + 255) / 256, 256, 0, stream>>>(i0p, winner, N, n);
  rs_gather_async<<<n, ROW_THREADS, 0, stream>>>(buffer, samples, winner, out);
}